// TerrellPenroseSNN_66443144069274
// MI455X (gfx1250) — compile-verified
//
#include <hip/hip_runtime.h>
#include <hip/hip_bf16.h>

#define BB   1024   // batch
#define IN   1024
#define HID  1024
#define OUT  512
#define TT   100

typedef __attribute__((ext_vector_type(2))) float v2f;
typedef __attribute__((ext_vector_type(8))) float v8f;

// ---------------------------------------------------------------------------
// Kernel 1: relativistic delay factors  delay[i] = exp(-gamma*|cd[i]|*v)
// (IN == HID == 1024, both slice causal_distances[:1024], so one vector.)
// ---------------------------------------------------------------------------
__global__ void k_delay(const float* __restrict__ cd,
                        const float* __restrict__ lf,
                        float* __restrict__ delay) {
    int i = blockIdx.x * blockDim.x + threadIdx.x;
    if (i >= IN) return;
    float v = lf[0];
    v = fminf(fmaxf(v, 0.0f), 0.999f);
    float gamma = 1.0f / sqrtf(1.0f - v * v);
    delay[i] = expf(-gamma * fabsf(cd[i]) * v);
}

// ---------------------------------------------------------------------------
// Kernel 2: w1d[i] = sum_h delay[h] * fc1_w[h, i]   (coalesced over i)
// ---------------------------------------------------------------------------
__global__ void k_w1d(const float* __restrict__ fc1_w,
                      const float* __restrict__ delay,
                      float* __restrict__ w1d) {
    int i = blockIdx.x * blockDim.x + threadIdx.x;
    if (i >= IN) return;
    float acc = 0.0f;
    for (int h = 0; h < HID; ++h)
        acc = fmaf(delay[h], fc1_w[(size_t)h * IN + i], acc);
    w1d[i] = acc;
}

// ---------------------------------------------------------------------------
// Kernel 3: c1 = dot(delay, fc1_b)   (single block of 1024)
// ---------------------------------------------------------------------------
__global__ __launch_bounds__(1024)
void k_c1(const float* __restrict__ delay,
          const float* __restrict__ fc1_b,
          float* __restrict__ c1) {
    __shared__ float red[32];
    int tid = threadIdx.x;
    float p = delay[tid] * fc1_b[tid];
    for (int o = 16; o; o >>= 1) p += __shfl_down(p, o, 32);
    if ((tid & 31) == 0) red[tid >> 5] = p;
    __syncthreads();
    if (tid == 0) {
        float s = 0.0f;
        for (int w = 0; w < 32; ++w) s += red[w];
        *c1 = s;
    }
}

// ---------------------------------------------------------------------------
// Kernel 4: rowsum2[o] = sum_h fc2_w[o, h]   (one block per row, coalesced)
// ---------------------------------------------------------------------------
__global__ __launch_bounds__(256)
void k_rowsum(const float* __restrict__ fc2_w,
              float* __restrict__ rowsum) {
    __shared__ float red[8];
    int o = blockIdx.x;
    int tid = threadIdx.x;
    float p = 0.0f;
    for (int h = tid; h < HID; h += 256) p += fc2_w[(size_t)o * HID + h];
    for (int s = 16; s; s >>= 1) p += __shfl_down(p, s, 32);
    if ((tid & 31) == 0) red[tid >> 5] = p;
    __syncthreads();
    if (tid == 0) {
        float s = 0.0f;
        for (int w = 0; w < 8; ++w) s += red[w];
        rowsum[o] = s;
    }
}

// ---------------------------------------------------------------------------
// Kernel 5 (the big one): drive[t,b] = sum_i x[b,t,i] * delay[i]
// via V_WMMA_F32_16X16X4_F32 tiles. One wave handles 16 rows of the
// flattened [B*T, IN] matrix. A-matrix lane layout (16x4 f32):
//   lane L: row M = L&15, K-pair = (L>>4)*2  -> one b64 load per K-chunk.
// B-matrix (4x16) has identical lane addressing with delay[] replicated
// across columns, so column 0 of D accumulates the 16 dot products.
// Bandwidth-bound: each cache line of x is fully consumed across the K loop.
// ---------------------------------------------------------------------------
__global__ __launch_bounds__(256)
void k_drive_wmma(const float* __restrict__ x,
                  const float* __restrict__ delay,
                  float* __restrict__ drive) {
    const int lane  = threadIdx.x & 31;
    const int wave  = blockIdx.x * (blockDim.x >> 5) + (threadIdx.x >> 5);
    const size_t rowBase = (size_t)wave * 16;       // rows r = b*T + t
    const int m  = lane & 15;                        // A row within tile
    const int kh = (lane >> 4) * 2;                  // K sub-offset: 0 or 2

    const float* arow = x + (rowBase + (size_t)m) * IN + kh;
    const float* brow = delay + kh;

    v8f c = {};
    #pragma unroll 4
    for (int k0 = 0; k0 < IN; k0 += 4) {
        v2f a = *(const v2f*)(arow + k0);            // 8B, 8B-aligned
        v2f b = *(const v2f*)(brow + k0);
        c = __builtin_amdgcn_wmma_f32_16x16x4_f32(
                /*neg_a=*/false, a, /*neg_b=*/false, b,
                /*c_mod=*/(short)0, c, /*reuse_a=*/false, /*reuse_b=*/false);
    }
    // D (16x16 f32): VGPR j -> lanes 0-15: M=j, N=lane; lanes 16-31: M=8+j.
    // Column N=0 lives in lane 0 (rows 0..7) and lane 16 (rows 8..15).
    if (lane == 0 || lane == 16) {
        const int mBase = (lane == 0) ? 0 : 8;
        #pragma unroll
        for (int j = 0; j < 8; ++j) {
            size_t r = rowBase + mBase + j;          // r = b*T + t
            int b = (int)(r / TT);
            int t = (int)(r % TT);
            drive[(size_t)t * BB + b] = c[j];
        }
    }
}

// ---------------------------------------------------------------------------
// Kernel 6: sequential LIF scan. One block, 1024 threads (thread == batch b).
// hm is uniform over B (scalar recurrence on thread 0). Epilogue writes out.
// ---------------------------------------------------------------------------
__global__ __launch_bounds__(1024)
void k_scan(const float* __restrict__ drive,
            const float* __restrict__ w1d,
            const float* __restrict__ c1p,
            const float* __restrict__ rowsum2,
            const float* __restrict__ fc2_b,
            const float* __restrict__ ta,
            const float* __restrict__ input_beta,
            const float* __restrict__ hidden_beta,
            const float* __restrict__ input_threshold,
            const float* __restrict__ hidden_threshold,
            float* __restrict__ out) {
    __shared__ float red[32];
    __shared__ float bc[2];
    const int b    = threadIdx.x;
    const int lane = b & 31;
    const int wv   = b >> 5;
    const float ib = input_beta[0];
    const float hb = hidden_beta[0];
    const float it = input_threshold[0];
    const float ht = hidden_threshold[0];
    const float wb = w1d[b];
    const float c1 = c1p[0];

    float im = 0.0f;
    float hm = 0.0f, accA = 0.0f, accT = 0.0f;   // thread-0 state

    for (int t = 0; t < TT; ++t) {
        im = im * ib + drive[(size_t)t * BB + b];
        float isp = (im > it) ? 1.0f : 0.0f;
        im *= (1.0f - isp);

        float p = isp * wb;                       // contribution to dot(w1d, isp)
        for (int o = 16; o; o >>= 1) p += __shfl_down(p, o, 32);
        if (lane == 0) red[wv] = p;
        __syncthreads();
        if (b == 0) {
            float s = c1;
            for (int w = 0; w < 32; ++w) s += red[w];
            hm = hm * hb + s;                     // uniform hidden membrane
            float hsp = (hm > ht) ? 1.0f : 0.0f;
            hm *= (1.0f - hsp);
            float tat = ta[t];
            accA += tat * hsp;
            accT += tat;
        }
        __syncthreads();                          // protect red[] for next step
    }
    if (b == 0) { bc[0] = accA; bc[1] = accT; }
    __syncthreads();
    if (b < OUT) out[b] = bc[0] * rowsum2[b] + bc[1] * fc2_b[b];
}

// ---------------------------------------------------------------------------
extern "C" void kernel_launch(void* const* d_in, const int* in_sizes, int n_in,
                              void* d_out, int out_size, void* d_ws, size_t ws_size,
                              hipStream_t stream) {
    const float* x    = (const float*)d_in[0];
    const float* cd   = (const float*)d_in[1];
    const float* fc1w = (const float*)d_in[2];
    const float* fc1b = (const float*)d_in[3];
    const float* fc2w = (const float*)d_in[4];
    const float* fc2b = (const float*)d_in[5];
    const float* ta   = (const float*)d_in[6];
    const float* lf   = (const float*)d_in[7];
    const float* ibeta = (const float*)d_in[8];
    const float* hbeta = (const float*)d_in[9];
    const float* ithr  = (const float*)d_in[10];
    const float* hthr  = (const float*)d_in[11];
    float* out = (float*)d_out;

    // workspace layout (floats); every element read is written first
    float* ws      = (float*)d_ws;
    float* delay   = ws;                 // 1024
    float* w1d     = ws + 1024;          // 1024
    float* rowsum2 = ws + 2048;          // 512
    float* c1      = ws + 2560;          // 1
    float* drive   = ws + 2576;          // 102400  (64B-aligned offset)

    k_delay <<<IN / 256, 256, 0, stream>>>(cd, lf, delay);
    k_w1d   <<<IN / 256, 256, 0, stream>>>(fc1w, delay, w1d);
    k_c1    <<<1, 1024, 0, stream>>>(delay, fc1b, c1);
    k_rowsum<<<OUT, 256, 0, stream>>>(fc2w, rowsum2);

    // (B*T)/16 = 6400 waves, 8 waves per 256-thread block -> 800 blocks
    k_drive_wmma<<<(BB * TT) / (16 * 8), 256, 0, stream>>>(x, delay, drive);

    k_scan  <<<1, 1024, 0, stream>>>(drive, w1d, c1, rowsum2, fc2b, ta,
                                     ibeta, hbeta, ithr, hthr, out);
}